// PConvBlock3D_11682311045742
// MI455X (gfx1250) — compile-verified
//
#include <hip/hip_runtime.h>
#include <hip/hip_bf16.h>

// ---- problem constants (from reference) ----
#define Bc    4
#define Cc    128
#define Dc    16
#define Hc    56
#define Wc    56
#define Kc    32          // selected channels
#define MIDc  32          // attention bottleneck
#define HWc   (Hc*Wc)     // 3136
#define DHWc  (Dc*HWc)    // 50176
#define BCDc  (Bc*Cc*Dc)  // 8192
#define TOTALc ((size_t)Bc*Cc*Dc*Hc*Wc)  // 25,690,112
#define REDc  (Kc*27)     // 864 reduction length for pconv GEMM
#define NTILE 8           // spatial tiles per block in pointwise GEMM
#define BN_EPS 1e-5f
#define NORM_EPS 1e-12f

typedef float v2f __attribute__((ext_vector_type(2)));
typedef float v8f __attribute__((ext_vector_type(8)));

// ---------------------------------------------------------------------------
// Kernel 1: per-(b,c,d) sum & sum-of-squares over H*W  -> pooled, norm
// ---------------------------------------------------------------------------
__global__ void stats_kernel(const float* __restrict__ x,
                             float* __restrict__ pooled,
                             float* __restrict__ nrm) {
    int bcd = blockIdx.x;                       // (b*C + c)*D + d
    const float* row = x + (size_t)bcd * HWc;
    float s = 0.f, ss = 0.f;
    for (int i = threadIdx.x; i < HWc; i += blockDim.x) {
        float v = row[i];
        s += v; ss += v * v;
    }
    __shared__ float ls[256], lss[256];
    ls[threadIdx.x] = s; lss[threadIdx.x] = ss;
    __syncthreads();
    for (int st = 128; st > 0; st >>= 1) {
        if ((int)threadIdx.x < st) {
            ls[threadIdx.x]  += ls[threadIdx.x + st];
            lss[threadIdx.x] += lss[threadIdx.x + st];
        }
        __syncthreads();
    }
    if (threadIdx.x == 0) {
        pooled[bcd] = ls[0] * (1.0f / HWc);
        nrm[bcd]    = fmaxf(sqrtf(lss[0]), NORM_EPS);
    }
}

// ---------------------------------------------------------------------------
// Kernel 2: meanv[b,d,n] = (1/C) * sum_c x[b,c,d,n] / norm[b,c,d]
// ---------------------------------------------------------------------------
__global__ void meanv_kernel(const float* __restrict__ x,
                             const float* __restrict__ nrm,
                             float* __restrict__ meanv) {
    int idx = blockIdx.x * blockDim.x + threadIdx.x;     // (b*D + d)*HW + n
    if (idx >= Bc * Dc * HWc) return;
    int n  = idx % HWc;
    int bd = idx / HWc;
    int d  = bd % Dc;
    int b  = bd / Dc;
    float acc = 0.f;
    for (int c = 0; c < Cc; ++c) {
        float v = x[(((size_t)(b * Cc + c) * Dc + d) * HWc) + n];
        acc += v / nrm[(b * Cc + c) * Dc + d];
    }
    meanv[idx] = acc * (1.0f / Cc);
}

// ---------------------------------------------------------------------------
// Kernel 3: sim[b,c,d] = (1/norm) * sum_n x[b,c,d,n] * meanv[b,d,n]
// ---------------------------------------------------------------------------
__global__ void sim_kernel(const float* __restrict__ x,
                           const float* __restrict__ nrm,
                           const float* __restrict__ meanv,
                           float* __restrict__ sim) {
    int bcd = blockIdx.x;
    int d   = bcd % Dc;
    int bc  = bcd / Dc;
    int b   = bc / Cc;
    const float* row = x + (size_t)bcd * HWc;
    const float* mv  = meanv + (size_t)(b * Dc + d) * HWc;
    float s = 0.f;
    for (int i = threadIdx.x; i < HWc; i += blockDim.x) s += row[i] * mv[i];
    __shared__ float ls[256];
    ls[threadIdx.x] = s;
    __syncthreads();
    for (int st = 128; st > 0; st >>= 1) {
        if ((int)threadIdx.x < st) ls[threadIdx.x] += ls[threadIdx.x + st];
        __syncthreads();
    }
    if (threadIdx.x == 0) sim[bcd] = ls[0] / nrm[bcd];
}

// ---------------------------------------------------------------------------
// Kernel 4: channel-attention MLP + importance + stable top-k(32) per (b,d)
// Rank-based selection matches jax.lax.top_k (desc value, ties: lower idx).
// ---------------------------------------------------------------------------
__global__ void topk_kernel(const float* __restrict__ pooled,
                            const float* __restrict__ sim,
                            const float* __restrict__ ca_w1,
                            const float* __restrict__ ca_b1,
                            const float* __restrict__ ca_w2,
                            const float* __restrict__ ca_b2,
                            const float* __restrict__ alpha_p,
                            int* __restrict__ sel) {
    int bd = blockIdx.x;
    int d  = bd % Dc;
    int b  = bd / Dc;
    int t  = threadIdx.x;                // 0..127 == channel index
    __shared__ float pool_s[Cc];
    __shared__ float h_s[MIDc];
    __shared__ float imp[Cc];

    pool_s[t] = pooled[(b * Cc + t) * Dc + d];
    __syncthreads();

    if (t < MIDc) {
        float a = ca_b1[t];
        for (int c = 0; c < Cc; ++c) a += ca_w1[t * Cc + c] * pool_s[c];
        h_s[t] = fmaxf(a, 0.f);
    }
    __syncthreads();

    float a2 = ca_b2[t];
    for (int m = 0; m < MIDc; ++m) a2 += ca_w2[t * MIDc + m] * h_s[m];
    float cw = 1.f / (1.f + __expf(-a2));
    float al = alpha_p[0];
    float s  = sim[(b * Cc + t) * Dc + d];
    imp[t] = al * (1.f - s) + (1.f - al) * cw;
    __syncthreads();

    float mine = imp[t];
    int rank = 0;
    for (int c = 0; c < Cc; ++c) {
        float v = imp[c];
        rank += (v > mine) || (v == mine && c < t);
    }
    if (rank < Kc) sel[bd * Kc + rank] = t;   // sel layout [B, D, K]
}

// ---------------------------------------------------------------------------
// Kernel 5: gathered 32->32 3x3x3 conv as WMMA GEMM, scattered into x_out.
// Reduction reordered tap-major: r' = tap*32 + kp, so every K=4 WMMA chunk
// stays inside one tap -> tap decode + depth/row bounds hoist to a uniform
// 27-iteration outer loop (skipped taps keep EXEC all-ones). Inner 8-chunk
// loop fully unrolled: 8 back-to-back v_wmma_f32_16x16x4_f32 per tap.
// Block = 64 threads (2 waves); wave w owns output-channel tile w.
// ---------------------------------------------------------------------------
__global__ void pconv_wmma_kernel(const float* __restrict__ x,
                                  const float* __restrict__ wp,   // [32,32,3,3,3]
                                  const int*   __restrict__ sel,  // [B,D,K]
                                  float* __restrict__ xout) {
    const int WT = 4;                         // ceil(56/16)
    int hw = blockIdx.x;
    int wt = hw % WT;
    int h  = hw / WT;
    int d  = blockIdx.y;
    int b  = blockIdx.z;
    int lane = threadIdx.x & 31;
    int wave = threadIdx.x >> 5;

    // selected channel table for depths d-1, d, d+1
    __shared__ int sel_s[3 * Kc];
    for (int i = threadIdx.x; i < 3 * Kc; i += blockDim.x) {
        int dz = i / Kc, kp = i % Kc;
        int dd = d + dz - 1;
        sel_s[i] = (dd >= 0 && dd < Dc) ? sel[(b * Dc + dd) * Kc + kp] : 0;
    }
    __syncthreads();

    int n  = lane & 15;        // column / M-row position within lane half
    int hi = lane >> 4;        // 0: K pair {0,1}, 1: K pair {2,3}
    int w0 = wt * 16 + n;      // spatial column (>=56 lanes load zero)
    int orow = wave * 16 + n;  // A-matrix row (output channel)

    v8f acc = {};
    for (int tap = 0; tap < 27; ++tap) {
        int dz  = tap / 9;
        int rem = tap % 9;
        int dy  = rem / 3;
        int dx  = rem % 3;
        int dd = d + dz - 1;
        int hh = h + dy - 1;
        if (dd < 0 || dd >= Dc || hh < 0 || hh >= Hc) continue;  // uniform skip
        int ww  = w0 + dx - 1;
        bool inw = (ww >= 0 && ww < Wc);
        const int* selrow = &sel_s[dz * Kc];
        const float* xslice = x + ((size_t)b * Cc * Dc + dd) * HWc
                                + (size_t)hh * Wc + ww;          // + c*D*HW later
        const float* wprow = wp + orow * REDc + tap;             // stride 27 per kp

        #pragma unroll
        for (int k = 0; k < Kc; k += 4) {
            int kp0 = k + hi * 2;
            v2f a, bb;
            a.x = wprow[kp0 * 27];
            a.y = wprow[(kp0 + 1) * 27];
            float b0 = 0.f, b1 = 0.f;
            if (inw) {
                int c0 = selrow[kp0];
                int c1 = selrow[kp0 + 1];
                b0 = xslice[(size_t)c0 * (Dc * HWc)];
                b1 = xslice[(size_t)c1 * (Dc * HWc)];
            }
            bb.x = b0; bb.y = b1;
            acc = __builtin_amdgcn_wmma_f32_16x16x4_f32(
                      false, a, false, bb, (short)0, acc, false, false);
        }
    }

    // scatter: conv output channel oc -> real channel sel[b, d, oc]
    int col = wt * 16 + n;
    if (col < Wc) {
        #pragma unroll
        for (int v = 0; v < 8; ++v) {
            int oc = wave * 16 + v + hi * 8;       // C/D layout: M = v + 8*hi
            int c  = sel_s[1 * Kc + oc];
            xout[(((size_t)(b * Cc + c) * Dc + d) * HWc) + h * Wc + col] = acc[v];
        }
    }
}

// ---------------------------------------------------------------------------
// Kernel 6: depthwise 3x3x3 conv + BN(eval) + ReLU, thread per output element
// ---------------------------------------------------------------------------
__global__ void dw_bn_relu_kernel(const float* __restrict__ xo,
                                  const float* __restrict__ dww,   // [C,1,3,3,3]
                                  const float* __restrict__ gamma,
                                  const float* __restrict__ beta,
                                  const float* __restrict__ mean,
                                  const float* __restrict__ var,
                                  float* __restrict__ y1) {
    size_t idx = (size_t)blockIdx.x * blockDim.x + threadIdx.x;
    if (idx >= TOTALc) return;
    int w = (int)(idx % Wc); size_t t = idx / Wc;
    int h = (int)(t % Hc); t /= Hc;
    int d = (int)(t % Dc); t /= Dc;
    int c = (int)(t % Cc);
    int b = (int)(t / Cc);
    const float* wc = dww + c * 27;
    float acc = 0.f;
    for (int dz = 0; dz < 3; ++dz) {
        int dd = d + dz - 1; if (dd < 0 || dd >= Dc) continue;
        for (int dy = 0; dy < 3; ++dy) {
            int hh = h + dy - 1; if (hh < 0 || hh >= Hc) continue;
            for (int dx = 0; dx < 3; ++dx) {
                int ww = w + dx - 1; if (ww < 0 || ww >= Wc) continue;
                acc += wc[dz * 9 + dy * 3 + dx] *
                       xo[(((size_t)(b * Cc + c) * Dc + dd) * HWc) + hh * Wc + ww];
            }
        }
    }
    float g = gamma[c] * rsqrtf(var[c] + BN_EPS);
    float v = (acc - mean[c]) * g + beta[c];
    y1[idx] = fmaxf(v, 0.f);
}

// ---------------------------------------------------------------------------
// Kernel 7: pointwise conv  out[b,o,s] = sum_c pw[o,c] * y1[b,c,s]  via WMMA.
// Block = 256 threads = 8 waves; wave w owns output-channel tile w (16 rows).
// Each block covers NTILE=8 spatial tiles; the wave's 64 A-operand scalars
// are preloaded to registers once (8x A-traffic reduction), then 8x32
// fully-unrolled v_wmma_f32_16x16x4_f32.
// Grid: x = DHW/(16*NTILE) = 392, y = b.  50176 = 392*128 exactly.
// ---------------------------------------------------------------------------
__global__ void pw_wmma_kernel(const float* __restrict__ y1,
                               const float* __restrict__ pw,   // [C,C]
                               float* __restrict__ out) {
    int tile0 = blockIdx.x * NTILE;
    int b     = blockIdx.y;
    int lane  = threadIdx.x & 31;
    int wave  = threadIdx.x >> 5;
    int n  = lane & 15;
    int hi = lane >> 4;
    int o  = wave * 16 + n;

    // preload this wave's A operands: for chunk k, lane needs pw[o, k+2*hi(+1)]
    float areg[2 * (Cc / 4)];
    #pragma unroll
    for (int k = 0; k < Cc; k += 4) {
        int c0 = k + hi * 2;
        areg[(k >> 2) * 2 + 0] = pw[o * Cc + c0];
        areg[(k >> 2) * 2 + 1] = pw[o * Cc + c0 + 1];
    }

    for (int t = 0; t < NTILE; ++t) {
        int s = (tile0 + t) * 16 + n;
        v8f acc = {};
        #pragma unroll
        for (int k = 0; k < Cc; k += 4) {
            int c0 = k + hi * 2;
            v2f a, bb;
            a.x = areg[(k >> 2) * 2 + 0];
            a.y = areg[(k >> 2) * 2 + 1];
            bb.x = y1[((size_t)(b * Cc + c0)) * DHWc + s];
            bb.y = y1[((size_t)(b * Cc + c0 + 1)) * DHWc + s];
            acc = __builtin_amdgcn_wmma_f32_16x16x4_f32(
                      false, a, false, bb, (short)0, acc, false, false);
        }
        #pragma unroll
        for (int v = 0; v < 8; ++v) {
            int oc = wave * 16 + v + hi * 8;
            out[((size_t)(b * Cc + oc)) * DHWc + s] = acc[v];
        }
    }
}

// ---------------------------------------------------------------------------
// Host launcher
// ---------------------------------------------------------------------------
extern "C" void kernel_launch(void* const* d_in, const int* in_sizes, int n_in,
                              void* d_out, int out_size, void* d_ws, size_t ws_size,
                              hipStream_t stream) {
    const float* x        = (const float*)d_in[0];
    const float* w_pconv  = (const float*)d_in[1];
    const float* ca_w1    = (const float*)d_in[2];
    const float* ca_b1    = (const float*)d_in[3];
    const float* ca_w2    = (const float*)d_in[4];
    const float* ca_b2    = (const float*)d_in[5];
    const float* alpha    = (const float*)d_in[6];
    const float* dw_w     = (const float*)d_in[7];
    const float* bn_gamma = (const float*)d_in[8];
    const float* bn_beta  = (const float*)d_in[9];
    const float* bn_mean  = (const float*)d_in[10];
    const float* bn_var   = (const float*)d_in[11];
    const float* pw_w     = (const float*)d_in[12];
    float* out = (float*)d_out;

    // workspace carve (256B aligned chunks)
    char* ws = (char*)d_ws;
    auto carve = [&](size_t bytes) {
        char* p = ws;
        ws += (bytes + 255) & ~(size_t)255;
        return p;
    };
    float* pooled = (float*)carve(sizeof(float) * BCDc);
    float* nrm    = (float*)carve(sizeof(float) * BCDc);
    float* sim    = (float*)carve(sizeof(float) * BCDc);
    int*   sel    = (int*)  carve(sizeof(int)   * Bc * Dc * Kc);
    float* meanv  = (float*)carve(sizeof(float) * Bc * Dc * HWc);
    float* xout   = (float*)carve(sizeof(float) * TOTALc);
    float* y1     = (float*)carve(sizeof(float) * TOTALc);

    // x_out starts as a copy of x; pconv scatters over selected channels
    hipMemcpyAsync(xout, x, sizeof(float) * TOTALc, hipMemcpyDeviceToDevice, stream);

    stats_kernel<<<BCDc, 256, 0, stream>>>(x, pooled, nrm);

    int bdn = Bc * Dc * HWc;
    meanv_kernel<<<(bdn + 255) / 256, 256, 0, stream>>>(x, nrm, meanv);

    sim_kernel<<<BCDc, 256, 0, stream>>>(x, nrm, meanv, sim);

    topk_kernel<<<Bc * Dc, 128, 0, stream>>>(pooled, sim, ca_w1, ca_b1,
                                             ca_w2, ca_b2, alpha, sel);

    pconv_wmma_kernel<<<dim3(Hc * 4, Dc, Bc), 64, 0, stream>>>(x, w_pconv, sel, xout);

    size_t total = TOTALc;
    dw_bn_relu_kernel<<<(unsigned)((total + 255) / 256), 256, 0, stream>>>(
        xout, dw_w, bn_gamma, bn_beta, bn_mean, bn_var, y1);

    pw_wmma_kernel<<<dim3(DHWc / (16 * NTILE), Bc), 256, 0, stream>>>(y1, pw_w, out);
}